// SiddonStep_90391881712195
// MI455X (gfx1250) — compile-verified
//
#include <hip/hip_runtime.h>
#include <math.h>

#define GRID_N      160
#define NVOX        (GRID_N * GRID_N * GRID_N)   // 4,096,000
#define NSTEPS      256
#define N_LORS      100000
#define WAVE        32
#define WAVES_PER_BLOCK 8
#define STEPS_PER_LANE  (NSTEPS / WAVE)          // 8
#define EPS_F       1e-8f

// TOF truncation: skip samples with exp argument < -43  (weight < ~5e-19,
// i.e. |s - s_tof| > 9 sigma). Contributions are ~16 orders of magnitude
// below the O(1) sums -> invisible in fp32, but saves ~half of the L2
// atomic/gather traffic that bounds this kernel.
#define EXP_CUT     (-43.0f)

// ---------------------------------------------------------------------------
// One wave32 per LOR. Lane l processes steps l, l+32, ..., l+224.
// compute (flat, w) in registers -> gather-MAC p -> wave32 butterfly
// reduction -> non-returning global_atomic_add_f32 into bp.
// ---------------------------------------------------------------------------
__global__ __launch_bounds__(WAVE * WAVES_PER_BLOCK)
void pet_fwd_bp_kernel(const float* __restrict__ image,
                       const float* __restrict__ lors,
                       float* __restrict__ bp)
{
    const int lane = threadIdx.x & (WAVE - 1);
    int lor = blockIdx.x * WAVES_PER_BLOCK + (threadIdx.x >> 5);
    // wave-uniform: force into SGPR so LOR descriptor loads scalarize
    lor = __builtin_amdgcn_readfirstlane(lor);

    const float* L7 = lors + (size_t)lor * 7;

    // Warm L2 for the LOR lines the next block will read
    // (gfx1250 global_prefetch_b8, speculative, no counter).
    if (lor + WAVES_PER_BLOCK < N_LORS)
        __builtin_prefetch(lors + (size_t)(lor + WAVES_PER_BLOCK) * 7, 0, 3);

    const float p0x = L7[0], p0y = L7[1], p0z = L7[2];
    const float p1x = L7[3], p1y = L7[4], p1z = L7[5];
    const float tof = L7[6];

    const float dx = p1x - p0x, dy = p1y - p0y, dz = p1z - p0z;
    const float L  = sqrtf(dx * dx + dy * dy + dz * dz);
    const float dl = L * (1.0f / NSTEPS);

    const float SIGMA     = 300.0f * 0.15f / 2.355f;   // TIME_RES*C_HALF/2.355
    const float inv_sigma = 1.0f / SIGMA;
    const float s_tof     = tof * 0.15f;

    int   flats[STEPS_PER_LANE];
    float ws[STEPS_PER_LANE];
    float p = 0.0f;

#pragma unroll
    for (int j = 0; j < STEPS_PER_LANE; ++j) {
        const int   step = lane + WAVE * j;
        const float t    = ((float)step + 0.5f) * (1.0f / NSTEPS);

        const float px = p0x + t * dx;
        const float py = p0y + t * dy;
        const float pz = p0z + t * dz;

        // idx = floor((p - ORIGIN)/VOXSIZE) = floor((p + 160) * 0.5)
        int ix = (int)floorf((px + 160.0f) * 0.5f);
        int iy = (int)floorf((py + 160.0f) * 0.5f);
        int iz = (int)floorf((pz + 160.0f) * 0.5f);

        const bool inb = (ix >= 0) & (ix < GRID_N) &
                         (iy >= 0) & (iy < GRID_N) &
                         (iz >= 0) & (iz < GRID_N);

        ix = min(max(ix, 0), GRID_N - 1);
        iy = min(max(iy, 0), GRID_N - 1);
        iz = min(max(iz, 0), GRID_N - 1);
        const int flat = ix * (GRID_N * GRID_N) + iy * GRID_N + iz;

        const float s = (t - 0.5f) * L;
        const float q = (s - s_tof) * inv_sigma;
        const float e = -0.5f * q * q;

        // test exponent BEFORE evaluating exp: dead lanes skip v_exp_f32,
        // the gather, and (below) the atomic, all via EXEC masking.
        float w = 0.0f;
        if (inb & (e > EXP_CUT))
            w = __expf(e) * dl;

        flats[j] = flat;
        ws[j]    = w;

        if (w != 0.0f)
            p += image[flat] * w;
    }

    // wave32 butterfly reduction of the forward projection p
#pragma unroll
    for (int off = WAVE / 2; off > 0; off >>= 1)
        p += __shfl_xor(p, off, WAVE);

    // backprojection: non-returning relaxed f32 atomics (STOREcnt path)
#pragma unroll
    for (int j = 0; j < STEPS_PER_LANE; ++j) {
        if (ws[j] != 0.0f) {
            const float v = p * ws[j];
            __hip_atomic_fetch_add(&bp[flats[j]], v,
                                   __ATOMIC_RELAXED, __HIP_MEMORY_SCOPE_AGENT);
        }
    }
}

// ---------------------------------------------------------------------------
// out[i] = image[i] / (effmap[i] + eps) * bp[i]   (in-place: out holds bp)
// float4-vectorized: 1,024,000 threads.
// ---------------------------------------------------------------------------
__global__ __launch_bounds__(256)
void pet_combine_kernel(const float* __restrict__ image,
                        const float* __restrict__ effmap,
                        float* __restrict__ out)
{
    const int i = blockIdx.x * blockDim.x + threadIdx.x;   // float4 index
    const float4 im = reinterpret_cast<const float4*>(image)[i];
    const float4 ef = reinterpret_cast<const float4*>(effmap)[i];
    float4 b        = reinterpret_cast<float4*>(out)[i];

    b.x = im.x / (ef.x + EPS_F) * b.x;
    b.y = im.y / (ef.y + EPS_F) * b.y;
    b.z = im.z / (ef.z + EPS_F) * b.z;
    b.w = im.w / (ef.w + EPS_F) * b.w;

    reinterpret_cast<float4*>(out)[i] = b;
}

extern "C" void kernel_launch(void* const* d_in, const int* in_sizes, int n_in,
                              void* d_out, int out_size, void* d_ws, size_t ws_size,
                              hipStream_t stream)
{
    (void)in_sizes; (void)n_in; (void)out_size; (void)d_ws; (void)ws_size;

    const float* image  = (const float*)d_in[0];
    const float* effmap = (const float*)d_in[1];
    const float* lors   = (const float*)d_in[2];
    float*       out    = (float*)d_out;

    // bp accumulates directly in d_out; zero it first (graph-capturable memset)
    hipMemsetAsync(out, 0, (size_t)NVOX * sizeof(float), stream);

    // 100,000 LORs / 8 waves-per-block = 12,500 blocks exactly
    pet_fwd_bp_kernel<<<N_LORS / WAVES_PER_BLOCK, WAVE * WAVES_PER_BLOCK, 0, stream>>>(
        image, lors, out);

    // 4,096,000 / 4 (float4) / 256 threads = 4,000 blocks exactly
    pet_combine_kernel<<<NVOX / 4 / 256, 256, 0, stream>>>(image, effmap, out);
}